// Net_14336600834598
// MI455X (gfx1250) — compile-verified
//
#include <hip/hip_runtime.h>
#include <hip/hip_bf16.h>
#include <math.h>

typedef float v2f __attribute__((ext_vector_type(2)));
typedef float v8f __attribute__((ext_vector_type(8)));

#define NPG   1024
#define NGR   100
#define NTOT  (NPG * NGR)
#define KNN   5

// ---------------------------------------------------------------------------
// Stage 1: per-node s = x@Ws^T+bs (3-dim), sn = |s|^2, h = x@Wh^T+bh (F-dim)
// ---------------------------------------------------------------------------
template <int CIN, int F>
__global__ __launch_bounds__(256) void precompute_sh(
    const float* __restrict__ x, const float* __restrict__ Ws,
    const float* __restrict__ bs, const float* __restrict__ Wh,
    const float* __restrict__ bh, float* __restrict__ s,
    float* __restrict__ sn, float* __restrict__ h) {
  int i = blockIdx.x * 256 + threadIdx.x;
  if (i >= NTOT) return;
  float xi[CIN];
#pragma unroll
  for (int c = 0; c < CIN; ++c) xi[c] = x[(size_t)i * CIN + c];
  float snv = 0.f;
#pragma unroll
  for (int r = 0; r < 3; ++r) {
    float a = bs[r];
#pragma unroll
    for (int c = 0; c < CIN; ++c) a = fmaf(Ws[r * CIN + c], xi[c], a);
    s[(size_t)i * 3 + r] = a;
    snv = fmaf(a, a, snv);
  }
  sn[i] = snv;
#pragma unroll
  for (int r = 0; r < F; ++r) {
    float a = bh[r];
#pragma unroll
    for (int c = 0; c < CIN; ++c) a = fmaf(Wh[r * CIN + c], xi[c], a);
    h[(size_t)i * F + r] = a;
  }
}

// ---------------------------------------------------------------------------
// Stage 2: fused kNN (WMMA f32 16x16x4 Gram tiles) + top-5 + weighted
// mean/max aggregation + output linear (Wo) + ReLU.
// grid = NGR*4 blocks of 256 threads (8 waves); each wave owns 32 nodes.
// ---------------------------------------------------------------------------
template <int CIN, int F, int COUT>
__global__ __launch_bounds__(256) void gravnet_knn_agg(
    const float* __restrict__ xin,  // [NTOT, CIN] layer input (for concat)
    const float* __restrict__ s,    // [NTOT, 3]
    const float* __restrict__ sn,   // [NTOT]
    const float* __restrict__ h,    // [NTOT, F]
    const float* __restrict__ Wo,   // [COUT, 2F+CIN]
    const float* __restrict__ bo,   // [COUT]
    float* __restrict__ out)        // [NTOT, COUT]
{
  constexpr int AGG = 2 * F + CIN;
  __shared__ float sh_s[NPG * 4];      // s0,s1,s2,sn interleaved: 16KB
  __shared__ float sh_d[8][32][18];    // per-wave d2 scratch (stride 18:
                                       // 8B-aligned rows, conflict-free)
  __shared__ float sh_w[COUT * AGG];   // Wo weights

  const int g     = blockIdx.x >> 2;
  const int blk   = blockIdx.x & 3;    // 256-node sub-block of the graph
  const int tid   = threadIdx.x;
  const int gbase = g * NPG;

  for (int i = tid; i < NPG; i += 256) {
    sh_s[i * 4 + 0] = s[(size_t)(gbase + i) * 3 + 0];
    sh_s[i * 4 + 1] = s[(size_t)(gbase + i) * 3 + 1];
    sh_s[i * 4 + 2] = s[(size_t)(gbase + i) * 3 + 2];
    sh_s[i * 4 + 3] = sn[gbase + i];
  }
  for (int i = tid; i < COUT * AGG; i += 256) sh_w[i] = Wo[i];
  __syncthreads();

  const int wave    = tid >> 5;
  const int lane    = tid & 31;
  const int hi      = lane >> 4;   // 0/1
  const int lm      = lane & 15;
  const int rowbase = blk * 256 + wave * 32;  // local node base of this wave
  // A/B fragment layout for V_WMMA_F32_16X16X4_F32:
  //   lane (hi,lm): A[M=lm][K=2*hi+v], B[K=2*hi+v][N=lm]; K=3 is zero pad.
  const float kmask = (hi == 0) ? 1.f : 0.f;   // K=1 real, K=3 padded
  const v2f* sh_s2  = (const v2f*)sh_s;        // k0=2*hi even -> 8B aligned

  // A fragments (loop-invariant): rows rowbase..+15 and rowbase+16..+31
  v2f aA = sh_s2[(rowbase + lm) * 2 + hi];
  v2f aB = sh_s2[(rowbase + 16 + lm) * 2 + hi];
  aA.y *= kmask;
  aB.y *= kmask;

  // loop-invariant sn_i for this lane's C-layout slots (regs survive the
  // asm memory clobber; avoids LDS reloads every tile)
  float sni0[8], sni1[8];
#pragma unroll
  for (int r = 0; r < 8; ++r) {
    const int m = r + 8 * hi;
    sni0[r] = sh_s[(rowbase + m) * 4 + 3];
    sni1[r] = sh_s[(rowbase + 16 + m) * 4 + 3];
  }

  // top-5 smallest d2 (ascending) per lane-owned node
  float bd[KNN] = {INFINITY, INFINITY, INFINITY, INFINITY, INFINITY};
  int   bi[KNN] = {0, 0, 0, 0, 0};

  // software-pipelined B fragment / sn_j for tile 0
  v2f   bcur   = sh_s2[lm * 2 + hi];
  float snjcur = sh_s[lm * 4 + 3];
  bcur.y *= kmask;

  for (int jt = 0; jt < NPG / 16; ++jt) {
    const int jb  = jt * 16;
    const v2f b   = bcur;
    const float snj = snjcur;

    v8f c0 = {0.f, 0.f, 0.f, 0.f, 0.f, 0.f, 0.f, 0.f};
    v8f c1 = {0.f, 0.f, 0.f, 0.f, 0.f, 0.f, 0.f, 0.f};
    c0 = __builtin_amdgcn_wmma_f32_16x16x4_f32(false, aA, false, b, (short)0,
                                               c0, false, false);
    c1 = __builtin_amdgcn_wmma_f32_16x16x4_f32(false, aB, false, b, (short)0,
                                               c1, false, false);

#pragma unroll
    for (int r = 0; r < 8; ++r) {
      const int m = r + 8 * hi;  // row within 16-tile; column n = lm
      sh_d[wave][m][lm]      = sni0[r] + snj - 2.f * c0[r];
      sh_d[wave][16 + m][lm] = sni1[r] + snj - 2.f * c1[r];
    }

    // prefetch next tile's B fragment (drained by the wait anyway => free)
    if (jt + 1 < NPG / 16) {
      bcur   = sh_s2[(jb + 16 + lm) * 2 + hi];
      snjcur = sh_s[(jb + 16 + lm) * 4 + 3];
      bcur.y *= kmask;
    }

    // wave-synchronous LDS transpose: other lanes' stores must land before
    // this lane reads its row (DS ops are in-order per wave; one wait does it)
    asm volatile("s_wait_dscnt 0" ::: "memory");

    // batch-read this lane's 16 d2 values (8 independent ds_load_b64),
    // then run the insertion entirely on registers
    v2f dv[8];
    const v2f* rowp = (const v2f*)&sh_d[wave][lane][0];
#pragma unroll
    for (int q = 0; q < 8; ++q) dv[q] = rowp[q];

#pragma unroll
    for (int n = 0; n < 16; ++n) {
      const float d = (n & 1) ? dv[n >> 1].y : dv[n >> 1].x;
      const int   j = jb + n;
      if (d < bd[4]) {
        if (d < bd[3]) { bd[4] = bd[3]; bi[4] = bi[3];
          if (d < bd[2]) { bd[3] = bd[2]; bi[3] = bi[2];
            if (d < bd[1]) { bd[2] = bd[1]; bi[2] = bi[1];
              if (d < bd[0]) { bd[1] = bd[0]; bi[1] = bi[0]; bd[0] = d; bi[0] = j; }
              else           { bd[1] = d; bi[1] = j; }
            } else { bd[2] = d; bi[2] = j; }
          } else { bd[3] = d; bi[3] = j; }
        } else { bd[4] = d; bi[4] = j; }
      }
    }
    // no trailing wait needed: per-wave DS ordering keeps next-tile stores
    // behind this tile's reads
  }

  // --- aggregation: w = exp(-10*max(d2,0)); weighted mean & max of h[j] ---
  const int node = rowbase + lane;          // local node
  const int gi   = gbase + node;            // global node
  float w[KNN];
#pragma unroll
  for (int k = 0; k < KNN; ++k) w[k] = __expf(-10.f * fmaxf(bd[k], 0.f));

  float mean[F], mx[F];
#pragma unroll
  for (int f = 0; f < F; ++f) { mean[f] = 0.f; mx[f] = -INFINITY; }
#pragma unroll
  for (int k = 0; k < KNN; ++k) {
    const float* hp = h + (size_t)(gbase + bi[k]) * F;
#pragma unroll
    for (int f = 0; f < F; ++f) {
      const float v = hp[f] * w[k];
      mean[f] += v;
      mx[f] = fmaxf(mx[f], v);
    }
  }
#pragma unroll
  for (int f = 0; f < F; ++f) mean[f] *= (1.f / KNN);

  float xi[CIN];
#pragma unroll
  for (int c = 0; c < CIN; ++c) xi[c] = xin[(size_t)gi * CIN + c];

#pragma unroll 4
  for (int c = 0; c < COUT; ++c) {
    const float* wr = sh_w + c * AGG;
    float a = bo[c];
#pragma unroll
    for (int f = 0; f < F; ++f) a = fmaf(wr[f], mean[f], a);
#pragma unroll
    for (int f = 0; f < F; ++f) a = fmaf(wr[F + f], mx[f], a);
#pragma unroll
    for (int ci = 0; ci < CIN; ++ci) a = fmaf(wr[2 * F + ci], xi[ci], a);
    out[(size_t)gi * COUT + c] = fmaxf(a, 0.f);
  }
}

// ---------------------------------------------------------------------------
// Stage 3: WMMA f32 GEMM  out[N,MOUT] = A[N,KC] @ W[MOUT,KC]^T + bias
// block = 256 (8 waves); each wave computes a 16-row stripe; W in LDS.
// ---------------------------------------------------------------------------
template <int KC, int MOUT, bool RELU>
__global__ __launch_bounds__(256) void gemm_wmma(
    const float* __restrict__ A, const float* __restrict__ W,
    const float* __restrict__ bias, float* __restrict__ out) {
  __shared__ float sh_w[MOUT * KC];
  __shared__ float sh_b[MOUT];
  for (int i = threadIdx.x; i < MOUT * KC; i += 256) sh_w[i] = W[i];
  for (int i = threadIdx.x; i < MOUT; i += 256) sh_b[i] = bias[i];
  __syncthreads();

  const int wave = threadIdx.x >> 5;
  const int lane = threadIdx.x & 31;
  const int hi   = lane >> 4;
  const int lm   = lane & 15;
  const int row0 = blockIdx.x * 128 + wave * 16;

  // preload this wave's A fragments (row = row0+lm, K chunk t: k = 4t+2*hi+v)
  v2f a[KC / 4];
  const float* arow = A + (size_t)(row0 + lm) * KC;
#pragma unroll
  for (int t = 0; t < KC / 4; ++t) {
    const int k = 4 * t + 2 * hi;
    a[t].x = arow[k];
    a[t].y = arow[k + 1];
  }

  constexpr int NT = (MOUT + 15) / 16;
#pragma unroll
  for (int nt = 0; nt < NT; ++nt) {
    const int  col = nt * 16 + lm;
    const bool cok = (col < MOUT);
    const float* wrow = sh_w + (cok ? col : 0) * KC;
    const float cmask = cok ? 1.f : 0.f;
    v8f c = {0.f, 0.f, 0.f, 0.f, 0.f, 0.f, 0.f, 0.f};
#pragma unroll
    for (int t = 0; t < KC / 4; ++t) {
      const int k = 4 * t + 2 * hi;
      v2f b;
      b.x = wrow[k] * cmask;
      b.y = wrow[k + 1] * cmask;
      c = __builtin_amdgcn_wmma_f32_16x16x4_f32(false, a[t], false, b, (short)0,
                                                c, false, false);
    }
    if (cok) {
      const float bb = sh_b[col];
#pragma unroll
      for (int r = 0; r < 8; ++r) {
        const int m = row0 + r + 8 * hi;
        float v = c[r] + bb;
        if (RELU) v = fmaxf(v, 0.f);
        out[(size_t)m * MOUT + col] = v;
      }
    }
  }
}

// ---------------------------------------------------------------------------
// Stage 4: per-graph pooling: relu(concat[max, min, sum, mean]) -> [G,496]
// ---------------------------------------------------------------------------
__global__ __launch_bounds__(128) void pool_kernel(const float* __restrict__ xg,
                                                   float* __restrict__ pooled) {
  const int g = blockIdx.x;
  const int f = threadIdx.x;
  if (f >= 124) return;
  float mx = -INFINITY, mn = INFINITY, sm = 0.f;
  const float* p = xg + (size_t)g * NPG * 124 + f;
  for (int n = 0; n < NPG; ++n) {
    const float v = p[(size_t)n * 124];
    mx = fmaxf(mx, v);
    mn = fminf(mn, v);
    sm += v;
  }
  float* o = pooled + g * 496;
  o[f]       = fmaxf(mx, 0.f);
  o[124 + f] = fmaxf(mn, 0.f);
  o[248 + f] = fmaxf(sm, 0.f);
  o[372 + f] = fmaxf(sm * (1.f / NPG), 0.f);
}

// ---------------------------------------------------------------------------
// Stage 5: per-graph MLP head: 496 -> 64 -> relu32 -> relu16 -> 1
// ---------------------------------------------------------------------------
__global__ __launch_bounds__(128) void head_kernel(
    const float* __restrict__ pooled, const float* __restrict__ W2,
    const float* __restrict__ b2, const float* __restrict__ Wg,
    const float* __restrict__ bg, const float* __restrict__ W3,
    const float* __restrict__ b3, const float* __restrict__ W4,
    const float* __restrict__ b4, float* __restrict__ out) {
  __shared__ float y0[496];
  __shared__ float y1[64];
  __shared__ float y2[32];
  __shared__ float y3[16];
  const int g = blockIdx.x;
  const int t = threadIdx.x;
  for (int i = t; i < 496; i += 128) y0[i] = pooled[g * 496 + i];
  __syncthreads();
  if (t < 64) {
    float a = b2[t];
    for (int k = 0; k < 496; ++k) a = fmaf(W2[t * 496 + k], y0[k], a);
    y1[t] = a;  // no relu here (matches reference)
  }
  __syncthreads();
  if (t < 32) {
    float a = bg[t];
    for (int k = 0; k < 64; ++k) a = fmaf(Wg[t * 64 + k], y1[k], a);
    y2[t] = fmaxf(a, 0.f);
  }
  __syncthreads();
  if (t < 16) {
    float a = b3[t];
    for (int k = 0; k < 32; ++k) a = fmaf(W3[t * 32 + k], y2[k], a);
    y3[t] = fmaxf(a, 0.f);
  }
  __syncthreads();
  if (t == 0) {
    float a = b4[0];
    for (int k = 0; k < 16; ++k) a = fmaf(W4[k], y3[k], a);
    out[g] = a;
  }
}

// ---------------------------------------------------------------------------
extern "C" void kernel_launch(void* const* d_in, const int* in_sizes, int n_in,
                              void* d_out, int out_size, void* d_ws,
                              size_t ws_size, hipStream_t stream) {
  const float* x   = (const float*)d_in[0];
  // d_in[1] = batch (regular, unused)
  const float* Ws1 = (const float*)d_in[2];
  const float* bs1 = (const float*)d_in[3];
  const float* Wh1 = (const float*)d_in[4];
  const float* bh1 = (const float*)d_in[5];
  const float* Wo1 = (const float*)d_in[6];
  const float* bo1 = (const float*)d_in[7];
  const float* Ws2 = (const float*)d_in[8];
  const float* bs2 = (const float*)d_in[9];
  const float* Wh2 = (const float*)d_in[10];
  const float* bh2 = (const float*)d_in[11];
  const float* Wo2 = (const float*)d_in[12];
  const float* bo2 = (const float*)d_in[13];
  const float* W1  = (const float*)d_in[14];
  const float* b1  = (const float*)d_in[15];
  const float* W2  = (const float*)d_in[16];
  const float* b2  = (const float*)d_in[17];
  const float* Wg  = (const float*)d_in[18];
  const float* bg  = (const float*)d_in[19];
  const float* W3  = (const float*)d_in[20];
  const float* b3  = (const float*)d_in[21];
  const float* W4  = (const float*)d_in[22];
  const float* b4  = (const float*)d_in[23];
  float* out = (float*)d_out;

  float* ws = (float*)d_ws;
  float* s      = ws;                         // NTOT*3
  float* sn     = s + (size_t)NTOT * 3;       // NTOT
  float* h      = sn + NTOT;                  // NTOT*32 (max F)
  float* x1     = h + (size_t)NTOT * 32;      // NTOT*32
  float* x2     = x1 + (size_t)NTOT * 32;     // NTOT*64
  float* x3     = x2 + (size_t)NTOT * 64;     // NTOT*124
  float* pooled = x3 + (size_t)NTOT * 124;    // NGR*496

  const int pblk = NTOT / 256;  // 400

  // Layer 1: GravNetConv(5 -> 32), space=3, prop=5
  precompute_sh<5, 5><<<pblk, 256, 0, stream>>>(x, Ws1, bs1, Wh1, bh1, s, sn, h);
  gravnet_knn_agg<5, 5, 32><<<NGR * 4, 256, 0, stream>>>(x, s, sn, h, Wo1, bo1, x1);

  // Layer 2: GravNetConv(32 -> 64), space=3, prop=32
  precompute_sh<32, 32><<<pblk, 256, 0, stream>>>(x1, Ws2, bs2, Wh2, bh2, s, sn, h);
  gravnet_knn_agg<32, 32, 64><<<NGR * 4, 256, 0, stream>>>(x1, s, sn, h, Wo2, bo2, x2);

  // W1: [NTOT,64] -> [NTOT,124] (no relu)
  gemm_wmma<64, 124, false><<<NTOT / 128, 256, 0, stream>>>(x2, W1, b1, x3);

  // Pool + head
  pool_kernel<<<NGR, 128, 0, stream>>>(x3, pooled);
  head_kernel<<<NGR, 128, 0, stream>>>(pooled, W2, b2, Wg, bg, W3, b3, W4, b4, out);
}